// DynamicDecayMemory_16071767622138
// MI455X (gfx1250) — compile-verified
//
#include <hip/hip_runtime.h>
#include <math.h>

// ---------------------------------------------------------------------------
// DynamicDecayMemory on MI455X (gfx1250, wave32, WMMA).
//
// Chunked (T=16) delta-rule reformulation:
//   per chunk:  P0 = M0 @ Kn^T (WMMA f32 16x16x4), G = Kn Kn^T (WMMA),
//               16-step scalar recurrence (pred_t = A_t p0_t - sum w_j G[j,t] u_j),
//               M <- A_end*M - sum_t w_t u_t kn_t^T  (rank-16 WMMA update).
// One workgroup per batch; M (256KB f32) resident in LDS (320KB/WGP).
// Cross-batch sn.max() per step: cluster barrier (-3) + agent-scope atomic
// counter barrier over a double-buffered sn exchange slab in d_ws.
// ---------------------------------------------------------------------------

#define DIMK 256
#define DIMV 256
#define BATCH 16
#define SEQ 4096
#define CH 16                    // chunk length == WMMA N
#define NCH (SEQ / CH)           // 256 chunks
#define LRf 0.1f
#define BASE_DECAY 0.01f
#define UWf 0.1f
#define SWf 0.1f
#define MAXD 0.5f
#define EPSF 1e-6f
#define MAXNORM 256.0f           // sqrt(256*256)

// LDS layout (floats), padded strides to avoid 16-way bank conflicts on the
// stride-256 WMMA operand reads (260 % 64 = 4 -> 16 distinct banks).
#define MSTR 260
#define KSTR 260
#define PSTR 17
#define GSTR 17
#define OFF_M   0
#define OFF_KN  (OFF_M  + DIMV * MSTR)   // 66560
#define OFF_U   (OFF_KN + CH * KSTR)     // 70720
#define OFF_P0  (OFF_U  + CH * DIMK)     // 74816
#define OFF_G   (OFF_P0 + DIMV * PSTR)   // 79168
#define OFF_W   (OFF_G  + CH * GSTR)     // 79440
#define OFF_SC  (OFF_W  + CH)            // 79456  [0]=A (decay product), [1]=||M||_F^2
#define OFF_RED (OFF_SC + 8)             // 79464
#define LDS_FLOATS (OFF_RED + 32)        // 79496 floats = 317,984 B  (< 320KB)

typedef float v2f __attribute__((ext_vector_type(2)));
typedef float v8f __attribute__((ext_vector_type(8)));

__device__ __forceinline__ v8f wmma4(v2f a, v2f b, v8f c) {
  // D(16x16,f32) = A(16x4,f32) * B(4x16,f32) + C
  return __builtin_amdgcn_wmma_f32_16x16x4_f32(false, a, false, b, (short)0, c,
                                               false, false);
}

__device__ __forceinline__ float waveReduceSum(float v) {
  // gfx1250 is wave32
  #pragma unroll
  for (int o = 16; o > 0; o >>= 1) v += __shfl_xor(v, o, 32);
  return v;
}

__global__ __launch_bounds__(256, 1)
void ddmem_kernel(const float* __restrict__ memIn,
                  const float* __restrict__ keys,
                  const float* __restrict__ values,
                  float* __restrict__ out,
                  unsigned* __restrict__ ctr,
                  float* __restrict__ snbuf) {
  extern __shared__ float lds[];

  const int b    = blockIdx.x;          // batch
  const int tid  = threadIdx.x;         // 0..255 (also the feature-dim index)
  const int wid  = tid >> 5;            // 8 waves
  const int lane = tid & 31;
  const int m    = lane & 15;           // WMMA row / col index within tile
  const int hi   = lane >> 4;           // half-wave select
  const int koff = hi * 2;              // K sub-offset for f32 A/B operands

  // ---- init: M -> LDS, accumulate ||M||_F^2 ------------------------------
  {
    float s = 0.f;
    const size_t mb = (size_t)b * DIMV * DIMK;
    for (int i = tid; i < DIMV * DIMK; i += 256) {
      float v = memIn[mb + i];
      lds[OFF_M + (i >> 8) * MSTR + (i & 255)] = v;
      s += v * v;
    }
    s = waveReduceSum(s);
    if (lane == 0) lds[OFF_RED + wid * 4] = s;
    __syncthreads();
    if (tid == 0) {
      float t = 0.f;
      for (int w = 0; w < 8; ++w) t += lds[OFF_RED + w * 4];
      lds[OFF_SC + 1] = t;              // normF2
    }
    __syncthreads();
  }

  float vreg[CH];   // this thread's v_t[tid] for the chunk
  float preg[CH];   // this thread's p0_t[tid]
  float ureg[CH];   // this thread's u_t[tid]

  for (int chunk = 0; chunk < NCH; ++chunk) {
    const int t0 = chunk * CH;

    // ---- phase A: load + normalize chunk keys into Kn, preload values ----
    #pragma unroll
    for (int rr = 0; rr < 2; ++rr) {
      const int t = wid * 2 + rr;
      const float* kp = keys + ((size_t)b * SEQ + t0 + t) * DIMK;
      float kv[8];
      float ss = 0.f;
      #pragma unroll
      for (int e = 0; e < 8; ++e) { kv[e] = kp[lane + e * 32]; ss += kv[e] * kv[e]; }
      ss = waveReduceSum(ss);
      const float scale = 1.0f / (sqrtf(ss) + EPSF);   // kn = k / (||k|| + eps)
      #pragma unroll
      for (int e = 0; e < 8; ++e)
        lds[OFF_KN + t * KSTR + lane + e * 32] = kv[e] * scale;
    }
    #pragma unroll
    for (int t = 0; t < CH; ++t)
      vreg[t] = values[((size_t)b * SEQ + t0 + t) * DIMV + tid];
    if (chunk + 1 < NCH) {   // warm L2/WGP$ for next chunk (global_prefetch_b8)
      __builtin_prefetch(keys   + ((size_t)b * SEQ + t0 + CH) * DIMK + tid * 16, 0, 1);
      __builtin_prefetch(values + ((size_t)b * SEQ + t0 + CH) * DIMV + tid * 16, 0, 1);
    }
    if (tid == 0) lds[OFF_SC + 0] = 1.0f;   // A (decay product) reset
    __syncthreads();

    // ---- phase A2 (WMMA): G = Kn Kn^T  (wave 0) --------------------------
    if (wid == 0) {
      v8f g = {};
      for (int k = 0; k < DIMK; k += 4) {
        v2f a = *(const v2f*)&lds[OFF_KN + m * KSTR + k + koff];
        g = wmma4(a, a, g);               // A and B operands are identical here
      }
      #pragma unroll
      for (int r = 0; r < 8; ++r)
        lds[OFF_G + (r + hi * 8) * GSTR + m] = g[r];
    }

    // ---- phase A3 (WMMA): P0 = M @ Kn^T  (256x16, 2 row-tiles per wave) --
    #pragma unroll
    for (int tt = 0; tt < 2; ++tt) {
      const int r0 = (wid + tt * 8) * 16;
      v8f acc = {};
      for (int k = 0; k < DIMK; k += 4) {
        v2f a  = *(const v2f*)&lds[OFF_M  + (r0 + m) * MSTR + k + koff];
        v2f bb = *(const v2f*)&lds[OFF_KN + m * KSTR        + k + koff];
        acc = wmma4(a, bb, acc);
      }
      #pragma unroll
      for (int r = 0; r < 8; ++r)
        lds[OFF_P0 + (r0 + r + hi * 8) * PSTR + m] = acc[r];
    }
    __syncthreads();
    #pragma unroll
    for (int t = 0; t < CH; ++t) preg[t] = lds[OFF_P0 + tid * PSTR + t];

    // ---- phase B: 16-step scalar/vector recurrence -----------------------
    for (int t = 0; t < CH; ++t) {
      const float Acur = lds[OFF_SC + 0];
      float pred = Acur * preg[t];
      for (int j = 0; j < t; ++j) {
        const float coef = lds[OFF_W + j] * lds[OFF_G + j * GSTR + t];
        pred -= coef * ureg[j];
      }
      const float vv  = vreg[t];
      const float sur = pred - vv;
      const float un  = 1.1f * pred - 0.1f * vv;   // pred + LR*surprise
      ureg[t] = un;
      lds[OFF_U + t * DIMK + tid] = un;            // needed by the update GEMM

      float s2 = waveReduceSum(sur * sur);
      float up = waveReduceSum(un * pred);
      float uu = waveReduceSum(un * un);
      if (lane == 0) {
        lds[OFF_RED + wid * 4 + 0] = s2;
        lds[OFF_RED + wid * 4 + 1] = up;
        lds[OFF_RED + wid * 4 + 2] = uu;
      }
      __syncthreads();

      const int gstep = chunk * CH + t;
      if (tid == 0) {
        float S2 = 0.f, UP = 0.f, UU = 0.f;
        for (int w = 0; w < 8; ++w) {
          S2 += lds[OFF_RED + w * 4 + 0];
          UP += lds[OFF_RED + w * 4 + 1];
          UU += lds[OFF_RED + w * 4 + 2];
        }
        const float sn  = sqrtf(S2);
        const int   par = gstep & 1;                 // double-buffered exchange
        __hip_atomic_store(&snbuf[par * BATCH + b], sn,
                           __ATOMIC_RELEASE, __HIP_MEMORY_SCOPE_AGENT);
        // Cluster barrier: one wave per WG signals + waits (-3). NOP when the
        // dispatch is not clustered; the atomic counter below is the portable
        // guarantee (16 co-resident WGs, one per WGP at 318KB LDS each).
        __builtin_amdgcn_s_cluster_barrier();
        __hip_atomic_fetch_add(ctr, 1u, __ATOMIC_RELEASE, __HIP_MEMORY_SCOPE_AGENT);
        const unsigned target = (unsigned)BATCH * (unsigned)(gstep + 1);
        while (__hip_atomic_load(ctr, __ATOMIC_ACQUIRE, __HIP_MEMORY_SCOPE_AGENT) < target)
          __builtin_amdgcn_s_sleep(1);
        float mx = 0.f;
        for (int bb2 = 0; bb2 < BATCH; ++bb2)
          mx = fmaxf(mx, __hip_atomic_load(&snbuf[par * BATCH + bb2],
                                           __ATOMIC_RELAXED, __HIP_MEMORY_SCOPE_AGENT));
        float normF2 = lds[OFF_SC + 1];
        const float util = fminf(fmaxf(sqrtf(normF2) / (MAXNORM + EPSF), 0.f), 1.f);
        const float snn  = sn / (mx + EPSF);
        float decay = BASE_DECAY * (1.f + UWf * util + SWf * snn);
        decay = fminf(fmaxf(decay, 0.f), MAXD);
        const float omd = 1.f - decay;
        // incremental Frobenius norm of the virtual M_t
        const float gtt = lds[OFF_G + t * GSTR + t];       // ||kn_t||^2
        normF2 = omd * omd * normF2 - 2.f * omd * UP + UU * gtt;
        lds[OFF_SC + 1] = fmaxf(normF2, 0.f);
        for (int j = 0; j < t; ++j) lds[OFF_W + j] *= omd;
        lds[OFF_W + t] = 1.f;
        lds[OFF_SC + 0] = Acur * omd;
      }
      __syncthreads();
    }

    // ---- phase C (WMMA): M = A_end*M - sum_t w_t * u_t kn_t^T ------------
    const float Aend = lds[OFF_SC + 0];
    for (int tl = wid; tl < 256; tl += 8) {      // 16x16 grid of 16x16 tiles
      const int r0 = (tl >> 4) * 16;
      const int c0 = (tl & 15) * 16;
      v8f acc;
      #pragma unroll
      for (int r = 0; r < 8; ++r)
        acc[r] = Aend * lds[OFF_M + (r0 + r + hi * 8) * MSTR + c0 + m];
      #pragma unroll
      for (int kb = 0; kb < CH; kb += 4) {
        const int kk = kb + koff;
        v2f a, bb;
        a.x  = -lds[OFF_W + kk]     * lds[OFF_U + kk * DIMK       + r0 + m];
        a.y  = -lds[OFF_W + kk + 1] * lds[OFF_U + (kk + 1) * DIMK + r0 + m];
        bb.x =  lds[OFF_KN + kk * KSTR       + c0 + m];
        bb.y =  lds[OFF_KN + (kk + 1) * KSTR + c0 + m];
        acc = wmma4(a, bb, acc);
      }
      #pragma unroll
      for (int r = 0; r < 8; ++r)
        lds[OFF_M + (r0 + r + hi * 8) * MSTR + c0 + m] = acc[r];
    }
    __syncthreads();
  }

  // ---- write final M -----------------------------------------------------
  {
    const size_t mb = (size_t)b * DIMV * DIMK;
    for (int i = tid; i < DIMV * DIMK; i += 256)
      out[mb + i] = lds[OFF_M + (i >> 8) * MSTR + (i & 255)];
  }
}

extern "C" void kernel_launch(void* const* d_in, const int* in_sizes, int n_in,
                              void* d_out, int out_size, void* d_ws, size_t ws_size,
                              hipStream_t stream) {
  const float* mem    = (const float*)d_in[0];  // (16,256,256)
  const float* keys   = (const float*)d_in[1];  // (16,4096,256)
  const float* values = (const float*)d_in[2];  // (16,4096,256)
  float* out = (float*)d_out;

  unsigned* ctr   = (unsigned*)d_ws;                     // barrier counter
  float*    snbuf = (float*)((char*)d_ws + 128);         // 2 x 16 sn slots

  // Reset inter-workgroup barrier state every launch (capture-safe).
  hipMemsetAsync(d_ws, 0, 256, stream);

  const size_t ldsBytes = (size_t)LDS_FLOATS * sizeof(float);  // 317,984 B
  ddmem_kernel<<<dim3(BATCH), dim3(256), ldsBytes, stream>>>(
      mem, keys, values, out, ctr, snbuf);
}